// ContrastiveLearninglossOnLatent_49409303773249
// MI455X (gfx1250) — compile-verified
//
#include <hip/hip_runtime.h>
#include <hip/hip_bf16.h>

typedef float v2f __attribute__((ext_vector_type(2)));
typedef float v8f __attribute__((ext_vector_type(8)));

#define B_ROWS 8192
#define ROW_STRIDE 512   // act row stride in floats (16*32); x uses first 256
#define TAUP 0.01f
#define TAUN 1.0f

// fragment read: rotate-swizzled column, invariant under row -> row+16
#define FRAG(buf, row, k) (*(const v2f*)(&buf[(row) * 64 + (((k) + 4 * (row)) & 63)]))

// ---------------------------------------------------------------------------
// Kernel 1: label histogram -> 1/c_pos, 1/c_neg into ws[0..1]; zero d_out.
// ---------------------------------------------------------------------------
__global__ __launch_bounds__(256) void init_kernel(const int* __restrict__ labels,
                                                   float* __restrict__ out,
                                                   float* __restrict__ wsf) {
  __shared__ int hist[8];
  int tid = threadIdx.x;
  if (tid < 8) hist[tid] = 0;
  __syncthreads();
  for (int i = tid; i < B_ROWS; i += 256) atomicAdd(&hist[labels[i] & 7], 1);
  __syncthreads();
  if (tid == 0) {
    double cpos = 0.0;
    for (int c = 0; c < 8; ++c) {
      double n = (double)hist[c];
      cpos += n * (n - 1.0) * 0.5;
    }
    double npairs = (double)B_ROWS * (double)(B_ROWS - 1) * 0.5;
    wsf[0] = (float)(1.0 / cpos);
    wsf[1] = (float)(1.0 / (npairs - cpos));
    out[0] = 0.0f;
  }
}

// ---------------------------------------------------------------------------
// Kernel 2: per-row squared norms. One wave32 per row, coalesced float2 loads.
// ---------------------------------------------------------------------------
__global__ __launch_bounds__(256) void sq_kernel(const float* __restrict__ act,
                                                 float* __restrict__ sqv) {
  int w = threadIdx.x >> 5;
  int lane = threadIdx.x & 31;
  int row = blockIdx.x * 8 + w;
  const float2* p = (const float2*)(act + (size_t)row * ROW_STRIDE);
  float s = 0.0f;
#pragma unroll
  for (int q = 0; q < 4; ++q) {         // 4 * 32 lanes * 2 floats = 256
    float2 v = p[q * 32 + lane];
    s += v.x * v.x + v.y * v.y;
  }
#pragma unroll
  for (int off = 16; off > 0; off >>= 1) s += __shfl_down(s, off, 32);
  if (lane == 0) sqv[row] = s;
}

// ---------------------------------------------------------------------------
// Kernel 3: 128x64 macrotile via V_WMMA_F32_16X16X4_F32.
// LDS tiles filled with GLOBAL_LOAD_ASYNC_TO_LDS_B128 in GVS mode
// (SGPR base + 32-bit byte offset), one s_wait_asynccnt per chunk.
// Each wave owns a 2x2 grid of 16x16 tiles: the paired fragment reads
// (rows 16 apart, swizzle-invariant) merge into ds_load_2addr_stride64_b64,
// giving 2 DS instructions per 4 WMMAs.
// ---------------------------------------------------------------------------
__global__ __launch_bounds__(256) void cl_tile_kernel(const float* __restrict__ act,
                                                      const int* __restrict__ labels,
                                                      const float* __restrict__ sqv,
                                                      const float* __restrict__ cnorm,
                                                      float* __restrict__ out) {
  const int bi = blockIdx.y;         // 128-row block, 0..63
  const int bj = blockIdx.x;         // 64-col block, 0..127
  if (bj < 2 * bi) return;           // whole block strictly below diagonal

  __shared__ float Ai[128 * 64];     // 32 KB, rotate-swizzled columns
  __shared__ float Bj[64 * 64];      // 16 KB
  __shared__ float sqa[128], sqb[64];
  __shared__ int   la[128], lb[64];
  __shared__ float red[256];

  const int tid  = threadIdx.x;
  const int w    = tid >> 5;         // wave 0..7
  const int lane = tid & 31;
  const int m    = lane & 15;
  const int hi   = lane >> 4;        // half-wave selector (K split in fragments)
  const int tr0  = (w & 3) * 2;      // wave's tile rows {tr0, tr0+1} of 8
  const int tc0  = (w >> 2) * 2;     // wave's tile cols {tc0, tc0+1} of 4

  // stage norms + labels for this macrotile
  if (tid < 128) {
    sqa[tid] = sqv[bi * 128 + tid];
    la[tid]  = labels[bi * 128 + tid] & 7;
  } else if (tid < 192) {
    int t = tid - 128;
    sqb[t] = sqv[bj * 64 + t];
    lb[t]  = labels[bj * 64 + t] & 7;
  }

  v8f acc[2][2] = {{{}, {}}, {{}, {}}};

  const int ra0 = tr0 * 16 + m;      // A rows this lane reads (pair 16 apart)
  const int ra1 = ra0 + 16;
  const int rb0 = tc0 * 16 + m;      // B "rows" (output columns), pair 16 apart
  const int rb1 = rb0 + 16;

  for (int kc = 0; kc < 4; ++kc) {   // K chunks of 64 (total K = 256)
    __syncthreads();                 // prior chunk fully consumed
    // ---- async DMA fills (GVS: s[base] + 32-bit byte offset) ----
#pragma unroll
    for (int p = 0; p < 8; ++p) {    // A tile: 128 rows x 16 float4
      int idx = tid + p * 256;
      int r   = idx >> 4;
      int c4  = idx & 15;
      unsigned goff = (unsigned)(((bi * 128 + r) * ROW_STRIDE + kc * 64 + c4 * 4) * 4);
      unsigned loff = (unsigned)(size_t)(&Ai[r * 64 + ((c4 * 4 + r * 4) & 63)]);
      asm volatile("global_load_async_to_lds_b128 %0, %1, %2"
                   :: "v"(loff), "v"(goff), "s"(act) : "memory");
    }
#pragma unroll
    for (int p = 0; p < 4; ++p) {    // B tile: 64 rows x 16 float4
      int idx = tid + p * 256;
      int r   = idx >> 4;
      int c4  = idx & 15;
      unsigned goff = (unsigned)(((bj * 64 + r) * ROW_STRIDE + kc * 64 + c4 * 4) * 4);
      unsigned loff = (unsigned)(size_t)(&Bj[r * 64 + ((c4 * 4 + r * 4) & 63)]);
      asm volatile("global_load_async_to_lds_b128 %0, %1, %2"
                   :: "v"(loff), "v"(goff), "s"(act) : "memory");
    }
    asm volatile("s_wait_asynccnt 0x0" ::: "memory");
    __syncthreads();

    // ---- WMMA phase: direct fragment reads (backend pipelines + merges) ----
#pragma unroll
    for (int k0 = 0; k0 < 64; k0 += 4) {
      int k = k0 + 2 * hi;           // per ISA 16x4 f32 fragment layout
      v2f a0 = FRAG(Ai, ra0, k);
      v2f a1 = FRAG(Ai, ra1, k);
      v2f b0 = FRAG(Bj, rb0, k);
      v2f b1 = FRAG(Bj, rb1, k);
      acc[0][0] = __builtin_amdgcn_wmma_f32_16x16x4_f32(false, a0, false, b0,
                                                        (short)0, acc[0][0], false, false);
      acc[0][1] = __builtin_amdgcn_wmma_f32_16x16x4_f32(false, a0, false, b1,
                                                        (short)0, acc[0][1], false, false);
      acc[1][0] = __builtin_amdgcn_wmma_f32_16x16x4_f32(false, a1, false, b0,
                                                        (short)0, acc[1][0], false, false);
      acc[1][1] = __builtin_amdgcn_wmma_f32_16x16x4_f32(false, a1, false, b1,
                                                        (short)0, acc[1][1], false, false);
    }
  }

  // ---- epilogue: mse -> contrastive terms (only i<j) -> block reduction ----
  const float inv_cpos = cnorm[0];
  const float inv_cneg = cnorm[1];
  float sum = 0.0f;

#pragma unroll
  for (int cc = 0; cc < 2; ++cc) {
    int jl = (tc0 + cc) * 16 + m;
    int jg = bj * 64 + jl;
    float sj = sqb[jl];
    int lj = lb[jl];
#pragma unroll
    for (int rr = 0; rr < 2; ++rr) {
#pragma unroll
      for (int v = 0; v < 8; ++v) {        // C/D layout: M = v + 8*hi
        int il = (tr0 + rr) * 16 + v + 8 * hi;
        int ig = bi * 128 + il;
        float mse = (sqa[il] + sj - 2.0f * acc[rr][cc][v]) * (1.0f / 256.0f);
        float c = (la[il] == lj) ? fmaxf(mse - TAUP, 0.0f) * inv_cpos
                                 : fmaxf(TAUN - mse, 0.0f) * inv_cneg;
        sum += (ig < jg) ? c : 0.0f;       // strict upper triangle == sum/2
      }
    }
  }

  red[tid] = sum;
  __syncthreads();
#pragma unroll
  for (int s = 128; s > 0; s >>= 1) {
    if (tid < s) red[tid] += red[tid + s];
    __syncthreads();
  }
  if (tid == 0) atomicAdd(out, red[0]);
}

// ---------------------------------------------------------------------------
extern "C" void kernel_launch(void* const* d_in, const int* in_sizes, int n_in,
                              void* d_out, int out_size, void* d_ws, size_t ws_size,
                              hipStream_t stream) {
  const float* act    = (const float*)d_in[0];
  const int*   labels = (const int*)d_in[1];
  float* out = (float*)d_out;
  float* wsf = (float*)d_ws;        // [0]=1/c_pos, [1]=1/c_neg
  float* sqv = wsf + 16;            // 8192 row norms

  init_kernel<<<1, 256, 0, stream>>>(labels, out, wsf);
  sq_kernel<<<B_ROWS / 8, 256, 0, stream>>>(act, sqv);
  dim3 grid(128, 64);               // 64-col blocks x 128-row blocks
  cl_tile_kernel<<<grid, 256, 0, stream>>>(act, labels, sqv, wsf, out);
}